// GATModel_50586124812430
// MI455X (gfx1250) — compile-verified
//
#include <hip/hip_runtime.h>
#include <hip/hip_bf16.h>
#include <stdint.h>
#include <stddef.h>

#define NUM_USERS 162541
#define NUM_ITEMS 59047
#define N_NODES   221588
#define DIM       128
#define N_EDGES   100000
#define E_TOTAL   (N_EDGES + N_NODES)   /* 321588 */
#define BATCH     16384
#define NEG_SLOPE 0.2f

#define SW_STRIDE 132   /* W rows in LDS: 128 + 4 pad DWORDs (TDM pad feature) */
#define SX_STRIDE 136   /* X tile rows in LDS: 128 + 8 pad floats             */
#define GEMM_MAX_BLOCKS 3072

typedef __attribute__((ext_vector_type(2))) float    v2f;
typedef __attribute__((ext_vector_type(4))) float    v4f;
typedef __attribute__((ext_vector_type(8))) float    v8f;
typedef __attribute__((ext_vector_type(4))) unsigned u32x4;
typedef __attribute__((ext_vector_type(4))) int      i32x4;
typedef __attribute__((ext_vector_type(8))) int      i32x8;

// ---------------------------------------------------------------------------
// monotone float <-> uint mapping so atomicMax(uint) implements float max
// ---------------------------------------------------------------------------
__device__ __forceinline__ unsigned flipf(float f) {
  unsigned u = __float_as_uint(f);
  return u ^ ((unsigned)((int)u >> 31) | 0x80000000u);
}
__device__ __forceinline__ float unflipf(unsigned b) {
  unsigned u = b ^ ((unsigned)((int)(~b) >> 31) | 0x80000000u);
  return __uint_as_float(u);
}

// ---------------------------------------------------------------------------
// 0) x = concat(user_table, item_table)            (float4 copies)
// ---------------------------------------------------------------------------
__global__ void concat_embed_k(const float* __restrict__ ut,
                               const float* __restrict__ it,
                               float* __restrict__ x) {
  size_t i = (size_t)blockIdx.x * blockDim.x + threadIdx.x;  // float4 units
  size_t total = (size_t)N_NODES * (DIM / 4);
  if (i >= total) return;
  size_t uc = (size_t)NUM_USERS * (DIM / 4);
  v4f v = (i < uc) ? ((const v4f*)ut)[i] : ((const v4f*)it)[i - uc];
  ((v4f*)x)[i] = v;
}

__global__ void zero_f32_k(float* __restrict__ p, size_t n) {
  size_t i = (size_t)blockIdx.x * blockDim.x + threadIdx.x;
  if (i < n) p[i] = 0.0f;
}

// ---------------------------------------------------------------------------
// 1) XW = X @ W   via V_WMMA_F32_16X16X4_F32, W staged by TDM, X tile by LDS.
//    Persistent blocks: each block TDM-loads W (128x128 f32) into LDS once
//    (with 4-DWORD row padding), then loops over 16-row tiles of X.
//    Fragment layouts (ISA 7.12.2, 32-bit):
//      A (16x4): lane<16 -> M=lane, K={0,1}; lane>=16 -> M=lane-16, K={2,3}
//      B (4x16): lane<16 -> N=lane, K={0,1}; lane>=16 -> N=lane-16, K={2,3}
//      C/D     : vgpr v, lanes0-15 -> M=v, lanes16-31 -> M=v+8; N=lane%16
// ---------------------------------------------------------------------------
__global__ void __launch_bounds__(256)
gemm_xw_wmma_k(const float* __restrict__ X, const float* __restrict__ W,
               float* __restrict__ XW, int nrows) {
  __shared__ float sW[128 * SW_STRIDE];   // 66 KB, filled by TDM
  __shared__ float sX[16 * SX_STRIDE];    // 8.5 KB, staged per tile

  const int tid  = threadIdx.x;
  const int wave = tid >> 5;              // 0..7 -> column tile of W
  const int lane = tid & 31;
  const int half = lane >> 4;
  const int l    = lane & 15;
  const int col  = wave * 16 + l;
  const int ntiles = (nrows + 15) >> 4;

  // ---- Tensor Data Mover: DMA W into sW (pad 4 DWORDs every 128 DWORDs) ----
  if (wave == 0) {
    unsigned lds_w = (unsigned)(uintptr_t)(void*)sW;   // LDS aperture: low 32b
    unsigned long long ga = (unsigned long long)(uintptr_t)W;
    u32x4 g0;
    g0[0] = 1u;                                        // count=1, user D#
    g0[1] = lds_w;                                     // lds_addr  [63:32]
    g0[2] = (unsigned)(ga & 0xffffffffull);            // global_addr[31:0]
    g0[3] = (unsigned)((ga >> 32) & 0x01ffffffull)     // global_addr[56:32]
            | (2u << 30);                              // type = 2 ("image")
    i32x8 g1;
    g1[0] = (2 << 16)        // data_size = 4 bytes
          | (1 << 20)        // pad_enable
          | (6 << 22)        // pad_interval: every 128 DWORDs
          | (3 << 25);       // pad_amount: 4 DWORDs
    g1[1] = (int)(128u << 16);     // tensor_dim0 low16 = 128
    g1[2] = (int)(128u << 16);     // tensor_dim1 low16 = 128
    g1[3] = (int)(128u << 16);     // tile_dim0 = 128
    g1[4] = 128;                   // tile_dim1 = 128, tile_dim2 = 0
    g1[5] = 128;                   // tensor_dim0_stride low32 = 128
    g1[6] = (int)(16384u << 16);   // tensor_dim1_stride low16 = 128*128
    g1[7] = 0;
    i32x4 gz = {0, 0, 0, 0};
#if __clang_major__ >= 23
    i32x8 gz8 = {0, 0, 0, 0, 0, 0, 0, 0};
    __builtin_amdgcn_tensor_load_to_lds(g0, g1, gz, gz, gz8, 0);
#else
    __builtin_amdgcn_tensor_load_to_lds(g0, g1, gz, gz, 0);
#endif
    __builtin_amdgcn_s_wait_tensorcnt(0);
  }

  for (int tile = blockIdx.x; tile < ntiles; tile += gridDim.x) {
    const int row_base = tile << 4;

    // cooperative stage of the 16x128 X tile (row-clamped, no divergence later)
    for (int chunk = tid; chunk < 512; chunk += 256) {   // 512 float4 chunks
      int r  = chunk >> 5;
      int cs = chunk & 31;
      int grow = row_base + r;
      if (grow >= nrows) grow = nrows - 1;               // clamp, store-guarded
      v4f v = *(const v4f*)(X + (size_t)grow * DIM + cs * 4);
      *(v4f*)(&sX[r * SX_STRIDE + cs * 4]) = v;
    }
    __syncthreads();   // also publishes sW on first iteration

    v8f acc = {0.f, 0.f, 0.f, 0.f, 0.f, 0.f, 0.f, 0.f};
#pragma unroll
    for (int k = 0; k < DIM; k += 4) {
      int kk = k + half * 2;
      v2f a = *(const v2f*)(&sX[l * SX_STRIDE + kk]);
      v2f b;
      b[0] = sW[kk * SW_STRIDE + col];
      b[1] = sW[(kk + 1) * SW_STRIDE + col];
      // 8 args: (neg_a, A, neg_b, B, c_mod, C, reuse_a, reuse_b)
      acc = __builtin_amdgcn_wmma_f32_16x16x4_f32(false, a, false, b,
                                                  (short)0, acc, false, false);
    }
#pragma unroll
    for (int v = 0; v < 8; v++) {
      int row = row_base + half * 8 + v;
      if (row < nrows) XW[(size_t)row * DIM + col] = acc[v];
    }
    __syncthreads();   // sX reused next iteration
  }
}

// ---------------------------------------------------------------------------
// 2) a_src[n,h] = <xw[n,h,:], att_src[h,:]>   (and same for dst)
// ---------------------------------------------------------------------------
__global__ void att_dots_k(const float* __restrict__ xw,
                           const float* __restrict__ aw_s,
                           const float* __restrict__ aw_d,
                           float* __restrict__ asrc, float* __restrict__ adst,
                           int H, int C) {
  int idx = blockIdx.x * blockDim.x + threadIdx.x;   // (n,h)
  if (idx >= N_NODES * H) return;
  int n = idx / H;
  int h = idx - n * H;
  const float* row = xw + (size_t)n * DIM + h * C;
  const float* ws = aw_s + h * C;
  const float* wd = aw_d + h * C;
  float s = 0.f, d = 0.f;
  for (int c = 0; c < C; c++) { float v = row[c]; s += v * ws[c]; d += v * wd[c]; }
  asrc[idx] = s;
  adst[idx] = d;
}

// ---------------------------------------------------------------------------
// 3) segment max of leaky_relu(a_src[src]+a_dst[dst]) over dst
// ---------------------------------------------------------------------------
__device__ __forceinline__ void edge_sd(const int* __restrict__ ei, int e,
                                        int& s, int& d) {
  if (e < N_EDGES) { s = ei[e]; d = ei[N_EDGES + e]; }
  else             { s = e - N_EDGES; d = s; }       // self-loop
}

__global__ void edge_max_k(const int* __restrict__ ei,
                           const float* __restrict__ asrc,
                           const float* __restrict__ adst,
                           unsigned* __restrict__ amax, int H) {
  int idx = blockIdx.x * blockDim.x + threadIdx.x;   // (e,h)
  if (idx >= E_TOTAL * H) return;
  int e = idx / H;
  int h = idx - e * H;
  int s, d; edge_sd(ei, e, s, d);
  float al = asrc[s * H + h] + adst[d * H + h];
  al = (al > 0.f) ? al : al * NEG_SLOPE;
  atomicMax(amax + (size_t)d * H + h, flipf(al));
}

// ---------------------------------------------------------------------------
// 4) ex = exp(alpha - amax[dst]);  denom[dst] += ex;  stash ex per edge
// ---------------------------------------------------------------------------
__global__ void edge_exp_k(const int* __restrict__ ei,
                           const float* __restrict__ asrc,
                           const float* __restrict__ adst,
                           const unsigned* __restrict__ amax,
                           float* __restrict__ denom,
                           float* __restrict__ exb, int H) {
  int idx = blockIdx.x * blockDim.x + threadIdx.x;
  if (idx >= E_TOTAL * H) return;
  int e = idx / H;
  int h = idx - e * H;
  int s, d; edge_sd(ei, e, s, d);
  float al = asrc[s * H + h] + adst[d * H + h];
  al = (al > 0.f) ? al : al * NEG_SLOPE;
  float ex = __expf(al - unflipf(amax[(size_t)d * H + h]));
  exb[idx] = ex;
  atomicAdd(denom + (size_t)d * H + h, ex);
}

// ---------------------------------------------------------------------------
// 5) out[dst,:] += att * xw[src,:]   (one wave per edge, 4 feats/lane)
// ---------------------------------------------------------------------------
__global__ void aggregate_k(const int* __restrict__ ei,
                            const float* __restrict__ xw,
                            const float* __restrict__ exb,
                            const float* __restrict__ denom,
                            float* __restrict__ outb, int H) {
  int e = (blockIdx.x * blockDim.x + threadIdx.x) >> 5;
  int lane = threadIdx.x & 31;
  if (e >= E_TOTAL) return;
  int s, d; edge_sd(ei, e, s, d);
  int C = DIM / H;
  const float* xs = xw + (size_t)s * DIM;
  float* od = outb + (size_t)d * DIM;
#pragma unroll
  for (int i = 0; i < 4; i++) {
    int f = lane + i * 32;
    int h = f / C;
    float att = exb[(size_t)e * H + h] / (denom[(size_t)d * H + h] + 1e-16f);
    atomicAdd(od + f, att * xs[f]);
  }
}

// ---------------------------------------------------------------------------
// 6) x_next = elu(out + bias)
// ---------------------------------------------------------------------------
__global__ void bias_elu_k(const float* __restrict__ inb,
                           const float* __restrict__ bias,
                           float* __restrict__ outb, size_t n) {
  size_t i = (size_t)blockIdx.x * blockDim.x + threadIdx.x;
  if (i >= n) return;
  float v = inb[i] + bias[i & (DIM - 1)];
  outb[i] = (v > 0.f) ? v : (__expf(v) - 1.f);
}

// ---------------------------------------------------------------------------
// 7) sigmoid(dot(x[user], x[item+NUM_USERS]))    one wave per batch element
// ---------------------------------------------------------------------------
__global__ void predict_k(const float* __restrict__ x,
                          const int* __restrict__ ui,
                          const int* __restrict__ ii,
                          float* __restrict__ outp) {
  int b = (blockIdx.x * blockDim.x + threadIdx.x) >> 5;
  int lane = threadIdx.x & 31;
  if (b >= BATCH) return;
  const float* u = x + (size_t)ui[b] * DIM;
  const float* v = x + ((size_t)ii[b] + NUM_USERS) * DIM;
  float s = 0.f;
#pragma unroll
  for (int i = 0; i < 4; i++) { int f = lane + i * 32; s += u[f] * v[f]; }
#pragma unroll
  for (int off = 16; off > 0; off >>= 1) s += __shfl_xor(s, off, 32);
  if (lane == 0) outp[b] = 1.f / (1.f + __expf(-s));
}

// ---------------------------------------------------------------------------
// host-side layer driver
// ---------------------------------------------------------------------------
static void run_gat_layer(const float* x, const float* W,
                          const float* att_src_w, const float* att_dst_w,
                          const float* bias, int H,
                          float* xw, float* outbuf, float* asrc, float* adst,
                          unsigned* amax, float* denom, float* exb,
                          float* xnext, const int* ei, hipStream_t stream) {
  int C = DIM / H;
  int row_tiles = (N_NODES + 15) / 16;
  int gemm_grid = row_tiles < GEMM_MAX_BLOCKS ? row_tiles : GEMM_MAX_BLOCKS;
  gemm_xw_wmma_k<<<gemm_grid, 256, 0, stream>>>(x, W, xw, N_NODES);

  int nh = N_NODES * H;
  att_dots_k<<<(nh + 255) / 256, 256, 0, stream>>>(xw, att_src_w, att_dst_w,
                                                   asrc, adst, H, C);
  // zero-init segment max (flip-space floor) and denom
  zero_f32_k<<<(nh + 255) / 256, 256, 0, stream>>>((float*)amax, (size_t)nh);
  zero_f32_k<<<(nh + 255) / 256, 256, 0, stream>>>(denom, (size_t)nh);

  int eh = E_TOTAL * H;
  edge_max_k<<<(eh + 255) / 256, 256, 0, stream>>>(ei, asrc, adst, amax, H);
  edge_exp_k<<<(eh + 255) / 256, 256, 0, stream>>>(ei, asrc, adst, amax, denom,
                                                   exb, H);
  size_t ne = (size_t)N_NODES * DIM;
  zero_f32_k<<<(unsigned)((ne + 255) / 256), 256, 0, stream>>>(outbuf, ne);
  aggregate_k<<<(E_TOTAL + 7) / 8, 256, 0, stream>>>(ei, xw, exb, denom,
                                                     outbuf, H);
  bias_elu_k<<<(unsigned)((ne + 255) / 256), 256, 0, stream>>>(outbuf, bias,
                                                               xnext, ne);
}

extern "C" void kernel_launch(void* const* d_in, const int* in_sizes, int n_in,
                              void* d_out, int out_size, void* d_ws,
                              size_t ws_size, hipStream_t stream) {
  (void)in_sizes; (void)n_in; (void)out_size; (void)ws_size;

  const int*   user_indices = (const int*)d_in[0];
  const int*   item_indices = (const int*)d_in[1];
  const int*   edge_index   = (const int*)d_in[2];
  const float* user_table   = (const float*)d_in[3];
  const float* item_table   = (const float*)d_in[4];
  const float* W1           = (const float*)d_in[5];
  const float* att_src1     = (const float*)d_in[6];
  const float* att_dst1     = (const float*)d_in[7];
  const float* bias1        = (const float*)d_in[8];
  const float* W2           = (const float*)d_in[9];
  const float* att_src2     = (const float*)d_in[10];
  const float* att_dst2     = (const float*)d_in[11];
  const float* bias2        = (const float*)d_in[12];
  float* out = (float*)d_out;

  // workspace carve-up (256B aligned slices)
  char* ws = (char*)d_ws;
  size_t off = 0;
  auto carve = [&](size_t bytes) -> char* {
    char* p = ws + off;
    off += (bytes + 255) & ~(size_t)255;
    return p;
  };
  const size_t node_feat_b = (size_t)N_NODES * DIM * sizeof(float);
  const size_t node_h_b    = (size_t)N_NODES * 4 * sizeof(float);     // H<=4
  const size_t edge_h_b    = (size_t)E_TOTAL * 4 * sizeof(float);     // H<=4

  float*    X     = (float*)carve(node_feat_b);
  float*    XW    = (float*)carve(node_feat_b);
  float*    OUT   = (float*)carve(node_feat_b);
  float*    ASRC  = (float*)carve(node_h_b);
  float*    ADST  = (float*)carve(node_h_b);
  unsigned* AMAX  = (unsigned*)carve(node_h_b);
  float*    DENOM = (float*)carve(node_h_b);
  float*    EXB   = (float*)carve(edge_h_b);

  // x = concat(user_table, item_table)
  size_t n4 = (size_t)N_NODES * (DIM / 4);
  concat_embed_k<<<(unsigned)((n4 + 255) / 256), 256, 0, stream>>>(
      user_table, item_table, X);

  // layer 1 (H=4), result written back into X
  run_gat_layer(X, W1, att_src1, att_dst1, bias1, /*H=*/4,
                XW, OUT, ASRC, ADST, AMAX, DENOM, EXB, X, edge_index, stream);
  // layer 2 (H=1), result written back into X
  run_gat_layer(X, W2, att_src2, att_dst2, bias2, /*H=*/1,
                XW, OUT, ASRC, ADST, AMAX, DENOM, EXB, X, edge_index, stream);

  // final link prediction
  predict_k<<<BATCH / 8, 256, 0, stream>>>(X, user_indices, item_indices, out);
}